// Attention_75342316306884
// MI455X (gfx1250) — compile-verified
//
#include <hip/hip_runtime.h>

// CDNA5 / gfx1250, wave32. fp32 WMMA (V_WMMA_F32_16X16X4_F32) + async-to-LDS DMA.

typedef __attribute__((ext_vector_type(2))) float v2f;
typedef __attribute__((ext_vector_type(8))) float v8f;

#define BATCH 64
#define CDIM  512
#define NDIM  784      // 28*28
#define ROWS  16       // channel rows per workgroup
#define RBS   788      // row-buffer LDS stride (bank-conflict-free A frags; RBS*4 % 16 == 0)
#define SBS   516      // logit/weight LDS stride
#define NTHREADS 256   // 8 wave32
#define SCALE 0.03571428571428571f   // 784^-0.5 = 1/28

__device__ __forceinline__ v8f wmma_f32(v2f a, v2f b, v8f c) {
    // D = A(16x4) * B(4x16) + C(16x16), fp32
    return __builtin_amdgcn_wmma_f32_16x16x4_f32(
        /*neg_a=*/false, a, /*neg_b=*/false, b,
        /*c_mod=*/(short)0, c, /*reuse_a=*/false, /*reuse_b=*/false);
}

__device__ __forceinline__ float sigmoidf(float v) {
    return 1.0f / (1.0f + __expf(-v));
}

// Async DMA: 16 rows x 784 floats global -> LDS, 16B granules, ASYNCcnt-tracked.
__device__ __forceinline__ void async_copy_rows(const float* __restrict__ src,
                                                float* rowbuf, int c0, int tid) {
    const int VPR = NDIM / 4;                       // 196 float4 per row
    for (int i = tid; i < ROWS * VPR; i += NTHREADS) {
        int r  = i / VPR;
        int c4 = (i - r * VPR) * 4;
        const float* gp = src + (size_t)(c0 + r) * NDIM + c4;
        unsigned lds = (unsigned)(unsigned long long)(uintptr_t)(rowbuf + r * RBS + c4);
        asm volatile("global_load_async_to_lds_b128 %0, %1, off"
                     :: "v"(lds), "v"(gp) : "memory");
    }
    asm volatile("s_wait_asynccnt 0" ::: "memory");
    __syncthreads();
}

// VALU path: load 16 rows with sigmoid applied in flight.
__device__ __forceinline__ void load_rows_sig(const float* __restrict__ src,
                                              float* rowbuf, int c0, int tid) {
    for (int i = tid; i < ROWS * NDIM; i += NTHREADS) {
        int r = i / NDIM;
        int c = i - r * NDIM;
        rowbuf[r * RBS + c] = sigmoidf(src[(size_t)(c0 + r) * NDIM + c]);
    }
    __syncthreads();
}

// Gram phase: Sbuf[m][d] = SCALE * sum_n rowbuf[m][n] * f(bsrc[d][n]),
// m in [0,16), d in [0,512). Each wave owns 64 columns (4 16x16 tiles).
template<bool SIG>
__device__ __forceinline__ void gram_phase(const float* __restrict__ bsrc,
                                           const float* rowbuf, float* Sbuf,
                                           int lane, int wave) {
    const int m  = lane & 15;
    const int ko = (lane >> 4) << 1;      // 0 or 2 (K sub-offset per half-wave)
    const int colBase = wave * 64;
    v8f acc0 = {}, acc1 = {}, acc2 = {}, acc3 = {};

    for (int k = 0; k < NDIM; k += 4) {
        // A fragment: 16x4 tile of the row block, from LDS (ds_load_b64)
        v2f a = *(const v2f*)(rowbuf + m * RBS + k + ko);

        // B fragments: B[k][d] = f(bsrc[d][k]) -> per-lane 8B global load
        v2f b0 = *(const v2f*)(bsrc + (size_t)(colBase +  0 + m) * NDIM + k + ko);
        v2f b1 = *(const v2f*)(bsrc + (size_t)(colBase + 16 + m) * NDIM + k + ko);
        v2f b2 = *(const v2f*)(bsrc + (size_t)(colBase + 32 + m) * NDIM + k + ko);
        v2f b3 = *(const v2f*)(bsrc + (size_t)(colBase + 48 + m) * NDIM + k + ko);
        if (SIG) {
            b0.x = sigmoidf(b0.x); b0.y = sigmoidf(b0.y);
            b1.x = sigmoidf(b1.x); b1.y = sigmoidf(b1.y);
            b2.x = sigmoidf(b2.x); b2.y = sigmoidf(b2.y);
            b3.x = sigmoidf(b3.x); b3.y = sigmoidf(b3.y);
        }
        acc0 = wmma_f32(a, b0, acc0);
        acc1 = wmma_f32(a, b1, acc1);
        acc2 = wmma_f32(a, b2, acc2);
        acc3 = wmma_f32(a, b3, acc3);
    }

    // Scatter accumulators (C/D layout: VGPR v -> M = v + 8*(lane>=16), N = lane&15)
    const int rowOff = (lane >> 4) << 3;
    #pragma unroll
    for (int v = 0; v < 8; ++v) {
        float* dst = Sbuf + (size_t)(v + rowOff) * SBS + colBase + m;
        dst[0]  = acc0[v] * SCALE;
        dst[16] = acc1[v] * SCALE;
        dst[32] = acc2[v] * SCALE;
        dst[48] = acc3[v] * SCALE;
    }
}

// Row softmax of Sbuf (16x512) in LDS; write or accumulate probs into Wbuf.
__device__ __forceinline__ void softmax_accum(float* Sbuf, float* Wbuf,
                                              float* red, float* rowstat,
                                              int tid, bool add) {
    const int r   = tid & 15;
    const int seg = tid >> 4;                 // 16 threads per row, 32 cols each
    float* srow = Sbuf + r * SBS + seg * 32;

    float mx = srow[0];
    #pragma unroll 4
    for (int j = 1; j < 32; ++j) mx = fmaxf(mx, srow[j]);
    red[r * 16 + seg] = mx;
    __syncthreads();
    if (seg == 0) {
        float m2 = red[r * 16];
        for (int j = 1; j < 16; ++j) m2 = fmaxf(m2, red[r * 16 + j]);
        rowstat[r] = m2;
    }
    __syncthreads();
    float rm = rowstat[r];
    float s = 0.0f;
    #pragma unroll 4
    for (int j = 0; j < 32; ++j) {
        float e = __expf(srow[j] - rm);
        srow[j] = e;
        s += e;
    }
    red[r * 16 + seg] = s;
    __syncthreads();
    if (seg == 0) {
        float t2 = 0.0f;
        for (int j = 0; j < 16; ++j) t2 += red[r * 16 + j];
        rowstat[r] = t2;
    }
    __syncthreads();
    float inv = 1.0f / rowstat[r];
    float* wrow = Wbuf + r * SBS + seg * 32;
    #pragma unroll 4
    for (int j = 0; j < 32; ++j) {
        float p = srow[j] * inv;
        wrow[j] = add ? (wrow[j] + p) : p;
    }
    __syncthreads();
}

__launch_bounds__(NTHREADS)
__global__ void attn_fused_kernel(const float* __restrict__ x,
                                  const float* __restrict__ cond,
                                  float* __restrict__ out) {
    __shared__ __align__(16) float rowbuf[ROWS * RBS];   // 50.4 KB
    __shared__ __align__(16) float Sbuf[ROWS * SBS];     // 33 KB logits
    __shared__ __align__(16) float Wbuf[ROWS * SBS];     // 33 KB weights
    __shared__ float red[ROWS * 16];
    __shared__ float rowstat[ROWS];

    const int tid  = threadIdx.x;
    const int lane = tid & 31;
    const int wave = tid >> 5;
    const int b    = blockIdx.y;
    const int c0   = blockIdx.x * ROWS;

    const float* xb = x    + (size_t)b * CDIM * NDIM;
    const float* cb = cond + (size_t)b * CDIM * NDIM;
    float*       ob = out  + (size_t)b * CDIM * NDIM;

    // P1-P3: w1 = softmax(scale * x_rows . x^T)
    async_copy_rows(xb, rowbuf, c0, tid);
    gram_phase<false>(xb, rowbuf, Sbuf, lane, wave);
    __syncthreads();
    softmax_accum(Sbuf, Wbuf, red, rowstat, tid, /*add=*/false);

    // P4-P6: w2 = softmax(scale * sig(c_rows) . sig(c)^T); W = w1 + w2
    load_rows_sig(cb, rowbuf, c0, tid);
    gram_phase<true>(cb, rowbuf, Sbuf, lane, wave);
    __syncthreads();
    softmax_accum(Sbuf, Wbuf, red, rowstat, tid, /*add=*/true);

    // P7: restore x rows for residual (DMA again; rowbuf was overwritten)
    async_copy_rows(xb, rowbuf, c0, tid);

    // P8: out_rows = x_rows + W(16x512) @ x(512x784)
    const int m  = lane & 15;
    const int ko = (lane >> 4) << 1;
    const int rowOff = (lane >> 4) << 3;
    for (int t = wave; t < 49; t += 8) {           // 49 column tiles of 16
        const int n0 = t * 16;
        v8f acc = {};
        for (int k = 0; k < CDIM; k += 4) {
            v2f a = *(const v2f*)(Wbuf + m * SBS + k + ko);
            v2f bv;
            bv.x = xb[(size_t)(k + ko    ) * NDIM + n0 + m];
            bv.y = xb[(size_t)(k + ko + 1) * NDIM + n0 + m];
            acc = wmma_f32(a, bv, acc);
        }
        #pragma unroll
        for (int v = 0; v < 8; ++v) {
            const int r = v + rowOff;
            const int n = n0 + m;
            ob[(size_t)(c0 + r) * NDIM + n] = rowbuf[r * RBS + n] + acc[v];
        }
    }
}

extern "C" void kernel_launch(void* const* d_in, const int* in_sizes, int n_in,
                              void* d_out, int out_size, void* d_ws, size_t ws_size,
                              hipStream_t stream) {
    (void)in_sizes; (void)n_in; (void)out_size; (void)d_ws; (void)ws_size;
    const float* x    = (const float*)d_in[0];
    const float* cond = (const float*)d_in[1];
    float*       out  = (float*)d_out;

    dim3 grid(CDIM / ROWS, BATCH);   // 32 row-blocks x 64 batches
    dim3 block(NTHREADS);
    attn_fused_kernel<<<grid, block, 0, stream>>>(x, cond, out);
}